// WeightedGraphSAGE_23381801959788
// MI455X (gfx1250) — compile-verified
//
#include <hip/hip_runtime.h>

typedef __attribute__((ext_vector_type(2))) float v2f;
typedef __attribute__((ext_vector_type(8))) float v8f;

// ---------------------------------------------------------------------------
// Edge scatter: one thread per (edge, 4-float feature chunk).
//   agg[dst] += w * h[src]   (per feature, global_atomic_add_f32, L2-resident)
//   deg[dst] += w            (chunk 0 only)
// Gathers are float4 -> global_load_b128; src/dst/w loads broadcast across
// the 2^shift consecutive threads sharing an edge.
// ---------------------------------------------------------------------------
__global__ void sage_scatter_edges(const float* __restrict__ h,
                                   const int* __restrict__ src,
                                   const int* __restrict__ dst,
                                   const float* __restrict__ w,
                                   float* __restrict__ agg,
                                   float* __restrict__ deg,
                                   long long total, int F, int shift) {
    long long tid = (long long)blockIdx.x * blockDim.x + threadIdx.x;
    if (tid >= total) return;
    int e = (int)(tid >> shift);
    int c = (int)(tid & ((1 << shift) - 1));   // float4 chunk within the row
    int s = src[e];
    int d = dst[e];
    float wt = w[e];
    const float4* hv = (const float4*)(h + (size_t)s * F);
    float4 v = hv[c];
    float* ap = agg + (size_t)d * F + c * 4;
    atomicAdd(ap + 0, v.x * wt);
    atomicAdd(ap + 1, v.y * wt);
    atomicAdd(ap + 2, v.z * wt);
    atomicAdd(ap + 3, v.w * wt);
    if (c == 0) atomicAdd(deg + d, wt);
}

// ---------------------------------------------------------------------------
// Fused SAGE dense stage, fp32 WMMA (V_WMMA_F32_16X16X4_F32).
//   NORM=true : A = (agg + hdst) / (deg + 1)  (normalization fused into loads)
//   NORM=false: A = hdst                      (plain GEMM for the FC head)
//   out = [relu](A @ W^T + bias),  W is (N,K) row-major.
// One wave computes one 16x16 output tile. K is processed in 32-wide blocks:
// phase 1 batch-loads 8 K-pairs into distinct registers (clause-able b64
// loads, one wait), phase 2 issues 8 WMMAs back-to-back into the matrix pipe.
// A-layout: lane L holds A[tm*16 + (L&15)][k0 + 2*(L>>4) + {0,1}]  (v2f)
// B-layout: lane L holds W[tn*16 + (L&15)][same K pair]            (v2f)
// C/D: VGPR r -> row tm*16 + r + 8*(L>=16), col tn*16 + (L&15).
// Requires K % 32 == 0 (K = 256 or 128 here).
// ---------------------------------------------------------------------------
template <bool NORM, bool RELU>
__global__ void sage_gemm_wmma(const float* __restrict__ agg,
                               const float* __restrict__ hdst,
                               const float* __restrict__ deg,
                               const float* __restrict__ W,
                               const float* __restrict__ bias,
                               float* __restrict__ out,
                               int K, int N, int ntilesN) {
    int tm = blockIdx.x;
    int tn = blockIdx.y * blockDim.y + threadIdx.y;   // wave-uniform
    if (tn >= ntilesN) return;                        // whole wave exits together

    int lane = threadIdx.x;        // 0..31
    int half = lane >> 4;          // 0: K pair {0,1}; 1: K pair {2,3}
    int l15  = lane & 15;
    int row  = tm * 16 + l15;      // A row for this lane
    int col  = tn * 16 + l15;      // B row (= output column) for this lane

    const v2f* arow2 = (const v2f*)((NORM ? agg : hdst) + (size_t)row * K);
    const v2f* hrow2 = (const v2f*)(hdst + (size_t)row * K);
    const v2f* brow2 = (const v2f*)(W + (size_t)col * K);
    float rinv = 1.0f;
    if (NORM) rinv = 1.0f / (deg[row] + 1.0f);

    v8f acc = {};
    for (int k0 = 0; k0 < K; k0 += 32) {
        v2f a[8], b[8];
        // Phase 1: batch loads (distinct regs -> load clause, single wait)
#pragma unroll
        for (int j = 0; j < 8; ++j) {
            int i = ((k0 + 4 * j) >> 1) + half;   // v2f index of lane's K pair
            if constexpr (NORM) {
                a[j] = (arow2[i] + hrow2[i]) * rinv;  // v_pk_add_f32 + v_pk_mul_f32
            } else {
                a[j] = arow2[i];
            }
            b[j] = brow2[i];
        }
        // Phase 2: 8 back-to-back WMMAs
#pragma unroll
        for (int j = 0; j < 8; ++j) {
            // (neg_a, A, neg_b, B, c_mod, C, reuse_a, reuse_b)
            acc = __builtin_amdgcn_wmma_f32_16x16x4_f32(
                false, a[j], false, b[j], (short)0, acc, false, false);
        }
    }

    float bcol = bias[col];
#pragma unroll
    for (int r = 0; r < 8; ++r) {
        int orow = tm * 16 + r + half * 8;
        float v = acc[r] + bcol;
        if constexpr (RELU) v = fmaxf(v, 0.0f);
        out[(size_t)orow * N + col] = v;
    }
}

// ---------------------------------------------------------------------------
// Problem constants (from the reference)
// ---------------------------------------------------------------------------
#define N0_SRC   495616
#define N0_DST   45056
#define E0       450560
#define N1_DST   4096
#define E1       40960
#define IN_FEATS 256
#define N_HIDDEN 128
#define N_CLASSES 16

extern "C" void kernel_launch(void* const* d_in, const int* in_sizes, int n_in,
                              void* d_out, int out_size, void* d_ws, size_t ws_size,
                              hipStream_t stream) {
    (void)in_sizes; (void)n_in; (void)out_size; (void)ws_size;
    const float* x    = (const float*)d_in[0];
    const int*   src0 = (const int*)  d_in[1];
    const int*   dst0 = (const int*)  d_in[2];
    const float* w0   = (const float*)d_in[3];
    const int*   src1 = (const int*)  d_in[4];
    const int*   dst1 = (const int*)  d_in[5];
    const float* w1   = (const float*)d_in[6];
    const float* Wn0  = (const float*)d_in[7];
    const float* bn0  = (const float*)d_in[8];
    const float* Wn1  = (const float*)d_in[9];
    const float* bn1  = (const float*)d_in[10];
    const float* Wfc  = (const float*)d_in[11];
    const float* bfc  = (const float*)d_in[12];
    float* out = (float*)d_out;

    // Workspace layout (floats): [agg0 | deg0 | agg1 | deg1 | h0 | h1]
    float* ws   = (float*)d_ws;
    float* agg0 = ws;                                      // 45056*256
    float* deg0 = agg0 + (size_t)N0_DST * IN_FEATS;        // 45056
    float* agg1 = deg0 + N0_DST;                           // 4096*128
    float* deg1 = agg1 + (size_t)N1_DST * N_HIDDEN;        // 4096
    float* h0   = deg1 + N1_DST;                           // 45056*128
    float* h1   = h0 + (size_t)N0_DST * N_HIDDEN;          // 4096*128

    // Zero the accumulators (agg0|deg0|agg1|deg1 are contiguous).
    size_t zero_floats = (size_t)N0_DST * IN_FEATS + N0_DST
                       + (size_t)N1_DST * N_HIDDEN + N1_DST;
    hipMemsetAsync(agg0, 0, zero_floats * sizeof(float), stream);

    // ---- Layer 0: scatter (E0 edges x 256 feats, 64 chunks/edge) ----
    {
        long long total = (long long)E0 << 6;
        int blocks = (int)((total + 255) / 256);
        sage_scatter_edges<<<blocks, 256, 0, stream>>>(
            x, src0, dst0, w0, agg0, deg0, total, IN_FEATS, 6);
    }
    // ---- Layer 0: fused normalize + GEMM + bias + relu -> h0 (45056x128) ----
    {
        dim3 blk(32, 4);
        dim3 grd(N0_DST / 16, (N_HIDDEN / 16 + 3) / 4);
        sage_gemm_wmma<true, true><<<grd, blk, 0, stream>>>(
            agg0, x, deg0, Wn0, bn0, h0, IN_FEATS, N_HIDDEN, N_HIDDEN / 16);
    }
    // ---- Layer 1: scatter (E1 edges x 128 feats, 32 chunks/edge) ----
    {
        long long total = (long long)E1 << 5;
        int blocks = (int)((total + 255) / 256);
        sage_scatter_edges<<<blocks, 256, 0, stream>>>(
            h0, src1, dst1, w1, agg1, deg1, total, N_HIDDEN, 5);
    }
    // ---- Layer 1: fused normalize + GEMM + bias + relu -> h1 (4096x128) ----
    {
        dim3 blk(32, 4);
        dim3 grd(N1_DST / 16, (N_HIDDEN / 16 + 3) / 4);
        sage_gemm_wmma<true, true><<<grd, blk, 0, stream>>>(
            agg1, h0, deg1, Wn1, bn1, h1, N_HIDDEN, N_HIDDEN, N_HIDDEN / 16);
    }
    // ---- FC head: out = h1 @ Wfc^T + bfc (4096x16), plain GEMM, no relu ----
    {
        dim3 blk(32, 4);
        dim3 grd(N1_DST / 16, 1);
        sage_gemm_wmma<false, false><<<grd, blk, 0, stream>>>(
            nullptr, h1, nullptr, Wfc, bfc, out, N_HIDDEN, N_CLASSES, 1);
    }
}